// Loss_37014028157101
// MI455X (gfx1250) — compile-verified
//
#include <hip/hip_runtime.h>
#include <hip/hip_bf16.h>

// ---------------------------------------------------------------------------
// Boundary loss for a 5-level point-cloud pyramid.
// Heavy part: brute-force KNN (self per level + cross vs level-0).
// Distance tiles computed on the CDNA5 matrix pipe with V_WMMA_F32_16X16X4_F32
// (K=4 holds the padded xyz coords), top-k selection kept on the VALU.
//
// KNN kernel shape (MI455X reasoning):
//  - one wave32 per workgroup  -> barrier ops are hardware NOPs (ISA 3.1),
//    so the LDS tile handoff between matrix and selection phases is ~free
//  - 32 candidates (two 16-wide tiles, 4 WMMAs) per loop iteration to
//    amortize loop/sync overhead and co-execute matrix ops with VALU top-k
//  - LDS tile rows padded to 33 floats: store phase spreads across banks,
//    scan phase stays contiguous (vectorizes to ds_load_b128)
// ---------------------------------------------------------------------------

typedef float v2f __attribute__((ext_vector_type(2)));
typedef float v8f __attribute__((ext_vector_type(8)));

#define KMAX      64
#define NUM_CLASSES 21

__global__ __launch_bounds__(32)
void knn_wmma_kernel(const float* __restrict__ Q,   // [B,S,3]
                     const float* __restrict__ X,   // [B,N,3]
                     int* __restrict__ out_idx,     // [B,S,k]
                     int S, int N, int k)
{
    __shared__ float qn_s[32];
    __shared__ float ds_s[32][33];     // [query row][candidate col], padded row

    const int lane   = threadIdx.x & 31;
    const int b      = blockIdx.y;
    const int q_base = blockIdx.x * 32;

    const float* __restrict__ Qb = Q + (size_t)b * S * 3;
    const float* __restrict__ Xb = X + (size_t)b * N * 3;

    const int hi   = lane >> 4;     // 0: lanes 0-15 carry K=0,1 ; 1: lanes 16-31 carry K=2,3
    const int lrow = lane & 15;

    // --- A fragments: rows q_base..q_base+15 (a0) and +16..+31 (a1) ---------
    v2f a0 = {0.f, 0.f}, a1 = {0.f, 0.f};
    const int r0 = q_base + lrow;
    const int r1 = q_base + 16 + lrow;
    if (r0 < S) {
        if (hi == 0) { a0.x = Qb[r0 * 3 + 0]; a0.y = Qb[r0 * 3 + 1]; }
        else         { a0.x = Qb[r0 * 3 + 2]; a0.y = 0.f; }
    }
    if (r1 < S) {
        if (hi == 0) { a1.x = Qb[r1 * 3 + 0]; a1.y = Qb[r1 * 3 + 1]; }
        else         { a1.x = Qb[r1 * 3 + 2]; a1.y = 0.f; }
    }

    // --- query norms for this wave's 32 rows --------------------------------
    const int qme = q_base + lane;
    float qn = 0.f;
    if (qme < S) {
        float x = Qb[qme * 3 + 0], y = Qb[qme * 3 + 1], z = Qb[qme * 3 + 2];
        qn = x * x + y * y + z * z;
    }
    qn_s[lane] = qn;
    __syncthreads();   // single-wave workgroup: S_NOP barrier + dscnt ordering

    // --- per-thread top-k (scratch arrays, ascending distances) -------------
    float best_d[KMAX];
    int   best_i[KMAX];
    for (int j = 0; j < k; ++j) { best_d[j] = 3.4e38f; best_i[j] = 0; }

    const bool qvalid = (qme < S);
    const int  rbase  = hi * 8;

    for (int ct = 0; ct < N; ct += 32) {
        // ---- candidate tile 0: columns ct .. ct+15 -------------------------
        {
            const int col = ct + lrow;
            const float cx = Xb[col * 3 + 0];
            const float cy = Xb[col * 3 + 1];
            const float cz = Xb[col * 3 + 2];
            v2f bf;
            bf.x = hi ? cz  : cx;
            bf.y = hi ? 0.f : cy;
            const float cn = cx * cx + cy * cy + cz * cz;

            v8f c0 = {}, c1 = {};
            v8f d0 = __builtin_amdgcn_wmma_f32_16x16x4_f32(false, a0, false, bf,
                                                           (short)0, c0, false, false);
            v8f d1 = __builtin_amdgcn_wmma_f32_16x16x4_f32(false, a1, false, bf,
                                                           (short)0, c1, false, false);
#pragma unroll
            for (int v = 0; v < 8; ++v) {
                const int tr = rbase + v;
                ds_s[tr][lrow]      = qn_s[tr]      + cn - 2.f * d0[v];
                ds_s[16 + tr][lrow] = qn_s[16 + tr] + cn - 2.f * d1[v];
            }
        }
        // ---- candidate tile 1: columns ct+16 .. ct+31 (uniform branch) -----
        if (ct + 16 < N) {
            const int col = ct + 16 + lrow;
            const float cx = Xb[col * 3 + 0];
            const float cy = Xb[col * 3 + 1];
            const float cz = Xb[col * 3 + 2];
            v2f bf;
            bf.x = hi ? cz  : cx;
            bf.y = hi ? 0.f : cy;
            const float cn = cx * cx + cy * cy + cz * cz;

            v8f c0 = {}, c1 = {};
            v8f d0 = __builtin_amdgcn_wmma_f32_16x16x4_f32(false, a0, false, bf,
                                                           (short)0, c0, false, false);
            v8f d1 = __builtin_amdgcn_wmma_f32_16x16x4_f32(false, a1, false, bf,
                                                           (short)0, c1, false, false);
#pragma unroll
            for (int v = 0; v < 8; ++v) {
                const int tr = rbase + v;
                ds_s[tr][16 + lrow]      = qn_s[tr]      + cn - 2.f * d0[v];
                ds_s[16 + tr][16 + lrow] = qn_s[16 + tr] + cn - 2.f * d1[v];
            }
        }
        __syncthreads();

        // ---- top-k maintenance: lane l owns query row l --------------------
        if (qvalid) {
            const int cc = (N - ct < 32) ? (N - ct) : 32;
            float kth = best_d[k - 1];
            for (int c = 0; c < cc; ++c) {
                const float d = ds_s[lane][c];
                if (d < kth) {
                    int pos = k - 1;
                    while (pos > 0 && best_d[pos - 1] > d) {   // strict: ties keep earlier index
                        best_d[pos] = best_d[pos - 1];
                        best_i[pos] = best_i[pos - 1];
                        --pos;
                    }
                    best_d[pos] = d;
                    best_i[pos] = ct + c;
                    kth = best_d[k - 1];
                }
            }
        }
        __syncthreads();
    }

    if (qvalid) {
        int* o = out_idx + ((size_t)b * S + qme) * k;
        for (int j = 0; j < k; ++j) o[j] = best_i[j];
    }
}

// ---------------------------------------------------------------------------
// Levels >= 1: probability_labels = mean of one-hot over k0 nearest base
// points  ->  center label = majority label (first max on tie == argmax).
// ---------------------------------------------------------------------------
__global__ void majority_label_kernel(const int* __restrict__ labels,     // [B,N0]
                                      const int* __restrict__ cross_idx,  // [B,S,k0]
                                      int* __restrict__ clabel,           // [B,S]
                                      int B, int S, int N0, int k0)
{
    const int t = blockIdx.x * blockDim.x + threadIdx.x;
    if (t >= B * S) return;
    const int b = t / S;
    const int p = t % S;

    int counts[NUM_CLASSES];
    for (int c = 0; c < NUM_CLASSES; ++c) counts[c] = 0;

    const int* nb = cross_idx + ((size_t)b * S + p) * k0;
    for (int j = 0; j < k0; ++j) {
        const int n = nb[j];
        counts[labels[(size_t)b * N0 + n]]++;
    }
    int best = -1, arg = 0;
    for (int c = 0; c < NUM_CLASSES; ++c)
        if (counts[c] > best) { best = counts[c]; arg = c; }
    clabel[t] = arg;
}

// ---------------------------------------------------------------------------
// Per-point boundary softmax contributions:
//   mask_j = (label[p] == label[nbr_j]); pm = 0 < cnt < ns
//   e_j = exp(-dist_j - max_j(-dist_j));  pos += e*mask, neg += e  (if pm)
// ---------------------------------------------------------------------------
__global__ void loss_points_kernel(const float* __restrict__ feat,     // [B,S,32]
                                   const int* __restrict__ self_idx,   // [B,S,ns]
                                   const int* __restrict__ clabel,     // [B,S]
                                   float* __restrict__ pos_acc,
                                   float* __restrict__ neg_acc,
                                   int* __restrict__ any_flag,
                                   int B, int S, int ns)
{
    const int t = blockIdx.x * blockDim.x + threadIdx.x;
    if (t >= B * S) return;
    const int b = t / S;
    const int p = t % S;

    const float* __restrict__ fb = feat + (size_t)b * S * 32;
    const float* __restrict__ fp = fb + (size_t)p * 32;
    float fc[32];
#pragma unroll
    for (int d = 0; d < 32; ++d) fc[d] = fp[d];

    const int* nb = self_idx + ((size_t)b * S + p) * ns;
    const int  cl = clabel[(size_t)b * S + p];

    float dv[KMAX];
    unsigned long long msk = 0ull;
    int   cnt    = 0;
    float negmax = -3.4e38f;

    for (int j = 0; j < ns; ++j) {
        const int n = nb[j];
        const float* __restrict__ fn = fb + (size_t)n * 32;
        float ss = 0.f;
#pragma unroll
        for (int d = 0; d < 32; ++d) {
            const float df = fc[d] - fn[d];
            ss = fmaf(df, df, ss);
        }
        const float dist = sqrtf(ss + 1e-6f);
        dv[j] = dist;
        if (clabel[(size_t)b * S + n] == cl) { ++cnt; msk |= (1ull << j); }
        negmax = fmaxf(negmax, -dist);
    }

    if (cnt > 0 && cnt < ns) {
        float ps = 0.f, ng = 0.f;
        for (int j = 0; j < ns; ++j) {
            const float e = expf(-dv[j] - negmax);   // TEMPERATURE == 1
            ng += e;
            if ((msk >> j) & 1ull) ps += e;
        }
        atomicAdd(pos_acc, ps);
        atomicAdd(neg_acc, ng);
        atomicOr(any_flag, 1);
    }
}

__global__ void init_accum_kernel(float* pos, float* neg, int* any)
{
    const int t = threadIdx.x;
    if (t < 5) { pos[t] = 0.f; neg[t] = 0.f; any[t] = 0; }
}

__global__ void finalize_kernel(const float* __restrict__ pos,
                                const float* __restrict__ neg,
                                const int* __restrict__ any,
                                float* __restrict__ out)
{
    if (threadIdx.x == 0 && blockIdx.x == 0) {
        float loss = 0.f;
        for (int i = 0; i < 5; ++i)
            if (any[i]) loss += -logf(pos[i] / (neg[i] + 1e-6f));
        out[0] = loss;
    }
}

// ---------------------------------------------------------------------------
// Host-side orchestration (graph-capture safe: all launches on `stream`,
// scratch carved out of d_ws, accumulators re-zeroed every call).
// Input dict order: labels, coord0, feat0, coord1, feat1, ..., coord4, feat4,
// num_classes.
// ---------------------------------------------------------------------------
extern "C" void kernel_launch(void* const* d_in, const int* in_sizes, int n_in,
                              void* d_out, int out_size, void* d_ws, size_t ws_size,
                              hipStream_t stream)
{
    (void)in_sizes; (void)n_in; (void)out_size; (void)ws_size;

    const int*   labels = (const int*)d_in[0];
    const float* coord[5] = { (const float*)d_in[1], (const float*)d_in[3],
                              (const float*)d_in[5], (const float*)d_in[7],
                              (const float*)d_in[9] };
    const float* feat[5]  = { (const float*)d_in[2], (const float*)d_in[4],
                              (const float*)d_in[6], (const float*)d_in[8],
                              (const float*)d_in[10] };

    const int B = 2;
    const int S[5]  = {4096, 1024, 256, 64, 16};
    const int NS[5] = {64, 32, 16, 8, 4};

    // workspace carving
    char*  ws  = (char*)d_ws;
    size_t off = 0;
    auto carve = [&](size_t bytes) -> void* {
        void* p = ws + off;
        off += (bytes + 255) & ~(size_t)255;
        return p;
    };

    int* clab[5] = {nullptr, nullptr, nullptr, nullptr, nullptr};
    for (int i = 1; i < 5; ++i) clab[i] = (int*)carve((size_t)B * S[i] * sizeof(int));
    int* sidx[5];
    for (int i = 0; i < 5; ++i) sidx[i] = (int*)carve((size_t)B * S[i] * NS[i] * sizeof(int));
    int* xidx[5] = {nullptr};
    for (int i = 1; i < 5; ++i) xidx[i] = (int*)carve((size_t)B * S[i] * (4 * i) * sizeof(int));
    float* pos = (float*)carve(5 * sizeof(float));
    float* neg = (float*)carve(5 * sizeof(float));
    int*   any = (int*)carve(5 * sizeof(int));

    init_accum_kernel<<<1, 32, 0, stream>>>(pos, neg, any);

    // cross-KNN vs base cloud + majority labels (levels 1..4)
    for (int i = 1; i < 5; ++i) {
        const int k0 = 4 * i;
        dim3 g((S[i] + 31) / 32, B);
        knn_wmma_kernel<<<g, 32, 0, stream>>>(coord[i], coord[0],
                                              xidx[i], S[i], S[0], k0);
        const int tot = B * S[i];
        majority_label_kernel<<<(tot + 127) / 128, 128, 0, stream>>>(
            labels, xidx[i], clab[i], B, S[i], S[0], k0);
    }

    // self-KNN + per-point loss terms (all levels)
    for (int i = 0; i < 5; ++i) {
        dim3 g((S[i] + 31) / 32, B);
        knn_wmma_kernel<<<g, 32, 0, stream>>>(coord[i], coord[i],
                                              sidx[i], S[i], S[i], NS[i]);
        const int* cl = (i == 0) ? labels : clab[i];
        const int tot = B * S[i];
        loss_points_kernel<<<(tot + 127) / 128, 128, 0, stream>>>(
            feat[i], sidx[i], cl, pos + i, neg + i, any + i, B, S[i], NS[i]);
    }

    finalize_kernel<<<1, 1, 0, stream>>>(pos, neg, any, (float*)d_out);
}